// OLD_TF_batch_poly_PSF_13039520710932
// MI455X (gfx1250) — compile-verified
//
#include <hip/hip_runtime.h>
#include <math.h>

typedef __attribute__((ext_vector_type(2))) float v2f;
typedef __attribute__((ext_vector_type(8))) float v8f;

#define N_BINS   8
#define BATCH_N  16
#define PUPIL    256
#define ODIM     64
#define TWO_PI   6.283185307179586f

__constant__ int d_phase_ns[N_BINS] = {1024, 988, 952, 918, 886, 856, 828, 802};

// D = A(16x4 f32) * B(4x16 f32) + C(16x16 f32), full-rate f32 matrix op on CDNA5
static __device__ __forceinline__ v8f wmma4(v2f a, v2f b, v8f c) {
  return __builtin_amdgcn_wmma_f32_16x16x4_f32(
      /*neg_a=*/false, a, /*neg_b=*/false, b,
      /*c_mod=*/(short)0, c, /*reuse_a=*/false, /*reuse_b=*/false);
}

// ---------------------------------------------------------------------------
// Kernel 1: per-bin pruned-DFT twiddles.
//   wtr/wti : [bin][k][x]  (64x256)  = cos / -sin of 2*pi*(k-32)*x/N_b
//   btr/bti : [bin][x][k]  (256x64)  = same values, transposed layout
// ---------------------------------------------------------------------------
__global__ void twiddle_kernel(float* __restrict__ wtr, float* __restrict__ wti,
                               float* __restrict__ btr, float* __restrict__ bti) {
  const int idx = blockIdx.x * blockDim.x + threadIdx.x;
  if (idx >= N_BINS * ODIM * PUPIL) return;
  const int bin = idx >> 14;          // / (64*256)
  const int rem = idx & 16383;
  const int k   = rem >> 8;           // 0..63  -> frequency k-32
  const int x   = rem & 255;          // 0..255
  const float N = (float)d_phase_ns[bin];
  const float t = TWO_PI * (float)((k - 32) * x) / N;
  float s, c;
  sincosf(t, &s, &c);
  wtr[idx] = c;
  wti[idx] = -s;
  btr[(bin << 14) + (x << 6) + k] = c;
  bti[(bin << 14) + (x << 6) + k] = -s;
}

// ---------------------------------------------------------------------------
// Kernel 2: one workgroup per (batch, bin).
//   Phase A: G(256x64, complex) = f * W^T ; f built on the fly from OPD.
//            G kept in dynamic LDS (128 KB).
//   Phase B: H(64x64, complex) = W * G ; psf = |H|^2 ; normalize ; weight.
// ---------------------------------------------------------------------------
__global__ __launch_bounds__(512, 1)
void psf_main_kernel(const float* __restrict__ opd,
                     const float* __restrict__ obsc,
                     const float* __restrict__ lambdas,
                     const float* __restrict__ sed,
                     const float* __restrict__ wtr, const float* __restrict__ wti,
                     const float* __restrict__ btr, const float* __restrict__ bti,
                     float* __restrict__ psfw) {
  extern __shared__ float smem[];
  float* Gr  = smem;                     // 256*64
  float* Gi  = smem + PUPIL * ODIM;      // 256*64
  float* red = smem + 2 * PUPIL * ODIM;  // 17 floats

  const int wg    = blockIdx.x;
  const int batch = wg >> 3;
  const int bin   = wg & 7;
  const int tid   = (int)threadIdx.x;
  const int wave  = tid >> 5;
  const int lane  = tid & 31;
  const int ln    = lane & 15;           // lane within half-wave
  const int hi    = lane >> 4;           // 0: K pair {0,1}, 1: K pair {2,3}

  const float phc = TWO_PI / lambdas[bin];
  const float* wR = wtr + bin * (ODIM * PUPIL);
  const float* wI = wti + bin * (ODIM * PUPIL);
  const float* bR = btr + bin * (PUPIL * ODIM);
  const float* bI = bti + bin * (PUPIL * ODIM);

  const v8f Z = {};

  // ----------------- Phase A: wave w owns G rows [16w, 16w+16), all 64 cols
  {
    const int row = wave * 16 + ln;                       // pupil row = A-frag M
    const float* oRow = opd + (batch * PUPIL + row) * PUPIL;
    const float* mRow = obsc + row * PUPIL;

    v8f P[4][4];
    #pragma unroll
    for (int nt = 0; nt < 4; ++nt)
      #pragma unroll
      for (int q = 0; q < 4; ++q) P[nt][q] = Z;

    for (int kb = 0; kb < PUPIL; kb += 4) {
      const int kx = kb + 2 * hi;
      const float o0 = oRow[kx], o1 = oRow[kx + 1];
      const float g0 = mRow[kx], g1 = mRow[kx + 1];
      float s0, c0, s1, c1;
      __sincosf(phc * o0, &s0, &c0);
      __sincosf(phc * o1, &s1, &c1);
      v2f ar; ar.x = g0 * c0; ar.y = g1 * c1;             // Re(f) fragment
      v2f ai; ai.x = g0 * s0; ai.y = g1 * s1;             // Im(f) fragment
      #pragma unroll
      for (int nt = 0; nt < 4; ++nt) {
        const int col = nt * 16 + ln;
        const float* pbr = bR + kx * ODIM + col;
        const float* pbi = bI + kx * ODIM + col;
        v2f br; br.x = pbr[0]; br.y = pbr[ODIM];          // B rows kx, kx+1
        v2f bi; bi.x = pbi[0]; bi.y = pbi[ODIM];
        P[nt][0] = wmma4(ar, br, P[nt][0]);               // Ar*Br
        P[nt][1] = wmma4(ai, bi, P[nt][1]);               // Ai*Bi
        P[nt][2] = wmma4(ar, bi, P[nt][2]);               // Ar*Bi
        P[nt][3] = wmma4(ai, br, P[nt][3]);               // Ai*Br
      }
    }
    #pragma unroll
    for (int nt = 0; nt < 4; ++nt) {
      const int n = nt * 16 + ln;
      #pragma unroll
      for (int r = 0; r < 8; ++r) {
        const int m = wave * 16 + r + 8 * hi;             // C/D row mapping
        Gr[m * ODIM + n] = P[nt][0][r] - P[nt][1][r];
        Gi[m * ODIM + n] = P[nt][2][r] + P[nt][3][r];
      }
    }
  }
  __syncthreads();

  // ----------------- Phase B: wave w owns H tile (mt = w/4, nt = w%4)
  const int mt = wave >> 2;
  const int nt = wave & 3;
  const int k1 = mt * 16 + ln;                            // H row = A-frag M
  const float* aR = wR + k1 * PUPIL;
  const float* aI = wI + k1 * PUPIL;
  const int n = nt * 16 + ln;                             // H col

  v8f Q0 = Z, Q1 = Z, Q2 = Z, Q3 = Z;
  for (int kb = 0; kb < PUPIL; kb += 4) {
    const int kx = kb + 2 * hi;
    v2f ar; ar.x = aR[kx];              ar.y = aR[kx + 1];
    v2f ai; ai.x = aI[kx];              ai.y = aI[kx + 1];
    v2f br; br.x = Gr[kx * ODIM + n];   br.y = Gr[(kx + 1) * ODIM + n];
    v2f bi; bi.x = Gi[kx * ODIM + n];   bi.y = Gi[(kx + 1) * ODIM + n];
    Q0 = wmma4(ar, br, Q0);
    Q1 = wmma4(ai, bi, Q1);
    Q2 = wmma4(ar, bi, Q2);
    Q3 = wmma4(ai, br, Q3);
  }

  float psf[8];
  float s = 0.0f;
  #pragma unroll
  for (int r = 0; r < 8; ++r) {
    const float hr = Q0[r] - Q1[r];
    const float hm = Q2[r] + Q3[r];
    psf[r] = hr * hr + hm * hm;
    s += psf[r];
  }
  // deterministic reduction: intra-wave shuffle tree, then wave0 over partials
  #pragma unroll
  for (int off = 16; off > 0; off >>= 1) s += __shfl_xor(s, off, 32);
  if (lane == 0) red[wave] = s;
  __syncthreads();
  if (wave == 0) {
    float t = (lane < 16) ? red[lane] : 0.0f;
    #pragma unroll
    for (int off = 16; off > 0; off >>= 1) t += __shfl_xor(t, off, 32);
    if (lane == 0) red[16] = t;
  }
  __syncthreads();

  const float scale = sed[bin] / red[16];
  float* outp = psfw + (batch * N_BINS + bin) * (ODIM * ODIM);
  #pragma unroll
  for (int r = 0; r < 8; ++r) {
    const int i = mt * 16 + r + 8 * hi;
    outp[i * ODIM + n] = psf[r] * scale;
  }
}

// ---------------------------------------------------------------------------
// Kernel 3: out[batch] = sum over bins of weighted normalized PSFs
// ---------------------------------------------------------------------------
__global__ void combine_kernel(const float* __restrict__ psfw, float* __restrict__ out) {
  const int idx = blockIdx.x * blockDim.x + threadIdx.x;
  if (idx >= BATCH_N * ODIM * ODIM) return;
  const int batch = idx >> 12;
  const int pix   = idx & 4095;
  const float* p = psfw + batch * N_BINS * 4096 + pix;
  float acc = 0.0f;
  #pragma unroll
  for (int b = 0; b < N_BINS; ++b) acc += p[b * 4096];
  out[idx] = acc;
}

extern "C" void kernel_launch(void* const* d_in, const int* in_sizes, int n_in,
                              void* d_out, int out_size, void* d_ws, size_t ws_size,
                              hipStream_t stream) {
  const float* opd     = (const float*)d_in[0];
  const float* obsc    = (const float*)d_in[1];
  const float* lambdas = (const float*)d_in[2];
  const float* sed     = (const float*)d_in[3];
  float* out = (float*)d_out;

  // workspace layout (floats): twiddles (512 KB x4 layouts) + per-bin PSFs (2 MB)
  float* ws   = (float*)d_ws;
  float* wtr  = ws;
  float* wti  = wtr + N_BINS * ODIM * PUPIL;
  float* btr  = wti + N_BINS * ODIM * PUPIL;
  float* bti  = btr + N_BINS * PUPIL * ODIM;
  float* psfw = bti + N_BINS * PUPIL * ODIM;   // 16*8*64*64 floats

  twiddle_kernel<<<(N_BINS * ODIM * PUPIL + 255) / 256, 256, 0, stream>>>(
      wtr, wti, btr, bti);

  const size_t smem = (size_t)(2 * PUPIL * ODIM + 32) * sizeof(float);  // ~128 KB
  (void)hipFuncSetAttribute(reinterpret_cast<const void*>(psf_main_kernel),
                            hipFuncAttributeMaxDynamicSharedMemorySize, (int)smem);
  psf_main_kernel<<<BATCH_N * N_BINS, 512, smem, stream>>>(
      opd, obsc, lambdas, sed, wtr, wti, btr, bti, psfw);

  combine_kernel<<<(BATCH_N * ODIM * ODIM + 255) / 256, 256, 0, stream>>>(psfw, out);
}